// CustomPrediction_30940944401003
// MI455X (gfx1250) — compile-verified
//
#include <hip/hip_runtime.h>
#include <cstdint>

typedef __attribute__((ext_vector_type(2))) float v2f;
typedef __attribute__((ext_vector_type(4))) float v4f;
typedef __attribute__((ext_vector_type(8))) float v8f;
typedef int i32x4 __attribute__((vector_size(16)));   // matches builtin's printed param type

#define TILE_M 128
#define TILE_N 128
#define TILE_K 32
#define A_STRIDE 36            // padded A LDS stride: conflict-free v2f frag reads, 16B-aligned fills
#define BP_COLS 144            // padded B pair-row: 2*144=288 words -> lane-half shift = 32 banks
#define BP_WORDS (16 * 2 * BP_COLS)

#define NEG_INF (-3.402823466e+38f)

#if __has_builtin(__builtin_amdgcn_global_load_async_to_lds_b128) && \
    __has_builtin(__builtin_amdgcn_global_load_async_to_lds_b32)
#define USE_ASYNC_LDS 1
#else
#define USE_ASYNC_LDS 0
#endif

#define AS3 __attribute__((address_space(3)))

#if USE_ASYNC_LDS
__device__ __forceinline__ void async_copy_b128(const void* g, void* l) {
    __builtin_amdgcn_global_load_async_to_lds_b128(
        (i32x4*)g,
        (AS3 i32x4*)(uint32_t)(uintptr_t)l,   // low 32 bits of flat LDS addr == LDS offset
        0, 0);
}
__device__ __forceinline__ void async_copy_b32(const void* g, void* l) {
    __builtin_amdgcn_global_load_async_to_lds_b32(
        (int*)g,
        (AS3 int*)(uint32_t)(uintptr_t)l,
        0, 0);
}
#endif

__device__ __forceinline__ void wait_async0() {
#if __has_builtin(__builtin_amdgcn_s_wait_asynccnt)
    __builtin_amdgcn_s_wait_asynccnt(0);
#else
    asm volatile("s_wait_asynccnt 0x0" ::: "memory");
#endif
}

#define WMMA_F32(a, b, c) \
    __builtin_amdgcn_wmma_f32_16x16x4_f32(false, (a), false, (b), (short)0, (c), false, false)

// ---------------------------------------------------------------------------
// f = X @ W in f32 via V_WMMA_F32_16X16X4_F32 (exact f32, preserves argmax).
// 256 threads (8 waves), block tile 128x128, wave tile 32x64.
// Double-buffered LDS; async global->LDS DMA staging when available.
// B is stored pair-interleaved in LDS so a B fragment {B[k][n],B[k+1][n]}
// is one contiguous ds_load_b64 (no register shuffles).
// ---------------------------------------------------------------------------
__global__ __launch_bounds__(256)
void gemm_f32_wmma(const float* __restrict__ A, const float* __restrict__ B,
                   float* __restrict__ C, int M, int N, int K)
{
    __shared__ float As[2][TILE_M][A_STRIDE];   // 2 x 18 KB
    __shared__ float Bp[2][BP_WORDS];           // 2 x 18 KB (pair-interleaved)

    const int tid   = threadIdx.x;
    const int wave  = tid >> 5;
    const int lane  = tid & 31;
    const int waveM = wave & 3;     // 4 waves along M (32 rows each)
    const int waveN = wave >> 2;    // 2 waves along N (64 cols each)
    const int bM    = blockIdx.x * TILE_M;
    const int bN    = blockIdx.y * TILE_N;
    const int lhalf = lane >> 4;
    const int l15   = lane & 15;

    v8f acc[2][4];
    #pragma unroll
    for (int i = 0; i < 2; ++i)
        #pragma unroll
        for (int j = 0; j < 4; ++j)
            acc[i][j] = {};

    const int nchunk = K / TILE_K;

    auto compute = [&](int buf) {
        #pragma unroll
        for (int kk = 0; kk < TILE_K; kk += 4) {
            const int kb = kk + lhalf * 2;   // lanes 0-15: K={kk,kk+1}; 16-31: K={kk+2,kk+3}
            v2f a0 = *(const v2f*)(&As[buf][waveM * 32 + l15][kb]);
            v2f a1 = *(const v2f*)(&As[buf][waveM * 32 + 16 + l15][kb]);
            const float* bprow = &Bp[buf][(kb >> 1) * (2 * BP_COLS)];  // kb even
            const int nb = waveN * 64 + l15;
            v2f b0 = *(const v2f*)(bprow + (nb)*2);
            v2f b1 = *(const v2f*)(bprow + (nb + 16) * 2);
            v2f b2 = *(const v2f*)(bprow + (nb + 32) * 2);
            v2f b3 = *(const v2f*)(bprow + (nb + 48) * 2);

            acc[0][0] = WMMA_F32(a0, b0, acc[0][0]);
            acc[0][1] = WMMA_F32(a0, b1, acc[0][1]);
            acc[0][2] = WMMA_F32(a0, b2, acc[0][2]);
            acc[0][3] = WMMA_F32(a0, b3, acc[0][3]);
            acc[1][0] = WMMA_F32(a1, b0, acc[1][0]);
            acc[1][1] = WMMA_F32(a1, b1, acc[1][1]);
            acc[1][2] = WMMA_F32(a1, b2, acc[1][2]);
            acc[1][3] = WMMA_F32(a1, b3, acc[1][3]);
        }
    };

#if USE_ASYNC_LDS
    // -------- async DMA staging (GLOBAL_LOAD_ASYNC_TO_LDS, ASYNCcnt) --------
    auto stage_async = [&](int kt, int buf) {
        // A tile 128x32: 4 contiguous b128 copies per thread
        #pragma unroll
        for (int j = 0; j < 4; ++j) {
            int idx = tid + 256 * j;
            int row = idx >> 3, c4 = idx & 7;
            async_copy_b128(A + (size_t)(bM + row) * K + kt + c4 * 4,
                            &As[buf][row][c4 * 4]);
        }
        // B tile 32x128: b32 copies scattered into pair-interleaved layout
        #pragma unroll
        for (int j = 0; j < 16; ++j) {
            int idx = tid + 256 * j;
            int k = idx >> 7, col = idx & 127;          // global side fully coalesced
            int w = (k >> 1) * (2 * BP_COLS) + col * 2 + (k & 1);
            async_copy_b32(B + (size_t)(kt + k) * N + bN + col,
                           &Bp[buf][w]);
        }
    };

    stage_async(0, 0);
    for (int t = 0; t < nchunk; ++t) {
        const int cur = t & 1;
        wait_async0();           // this wave's DMA for chunk t done
        __syncthreads();         // everyone's DMA done; prior readers of buf cur^1 done
        if (t + 1 < nchunk) stage_async((t + 1) * TILE_K, cur ^ 1);  // overlaps compute
        compute(cur);
    }
#else
    // -------- fallback: register-staged double buffering --------
    v4f ra[4], rb[4];
    auto gload = [&](int kt) {
        #pragma unroll
        for (int j = 0; j < 4; ++j) {
            int idx = tid + 256 * j;
            int row = idx >> 3, c4 = idx & 7;
            ra[j] = *(const v4f*)(A + (size_t)(bM + row) * K + kt + c4 * 4);
        }
        #pragma unroll
        for (int j = 0; j < 4; ++j) {
            int idx = tid + 256 * j;
            int row = idx >> 5, c4 = idx & 31;
            rb[j] = *(const v4f*)(B + (size_t)(kt + row) * N + bN + c4 * 4);
        }
    };
    auto lstore = [&](int buf) {
        #pragma unroll
        for (int j = 0; j < 4; ++j) {
            int idx = tid + 256 * j;
            int row = idx >> 3, c4 = idx & 7;
            *(v4f*)(&As[buf][row][c4 * 4]) = ra[j];
        }
        #pragma unroll
        for (int j = 0; j < 4; ++j) {
            int idx = tid + 256 * j;
            int row = idx >> 5, c4 = idx & 31;
            int base = (row >> 1) * (2 * BP_COLS) + (c4 * 4) * 2 + (row & 1);
            Bp[buf][base + 0] = rb[j].x;
            Bp[buf][base + 2] = rb[j].y;
            Bp[buf][base + 4] = rb[j].z;
            Bp[buf][base + 6] = rb[j].w;
        }
    };

    gload(0);
    lstore(0);
    for (int t = 0; t < nchunk; ++t) {
        const int cur = t & 1;
        __syncthreads();                         // buf[cur] visible; prior readers done
        if (t + 1 < nchunk) gload((t + 1) * TILE_K);
        compute(cur);
        if (t + 1 < nchunk) lstore(cur ^ 1);
    }
#endif

    // ---- epilogue: VGPR r holds M=r (lanes 0-15) / M=r+8 (lanes 16-31) ----
    #pragma unroll
    for (int mi = 0; mi < 2; ++mi) {
        #pragma unroll
        for (int ni = 0; ni < 4; ++ni) {
            const int n     = bN + waveN * 64 + ni * 16 + l15;
            const int mbase = bM + waveM * 32 + mi * 16 + lhalf * 8;
            #pragma unroll
            for (int r = 0; r < 8; ++r)
                C[(size_t)(mbase + r) * N + n] = acc[mi][ni][r];
        }
    }
}

// ---------------------------------------------------------------------------
// Tree descent: one block per batch row. 256 threads = 16 candidates x 16
// d-groups. Candidates of a node are consecutive Xi columns -> at fixed d the
// 16 gathers form one 64B cacheline (coalesced without transposing Xi).
// ---------------------------------------------------------------------------
__global__ __launch_bounds__(256)
void tree_descend(const float* __restrict__ f, const float* __restrict__ Xi,
                  const int* __restrict__ children, int* __restrict__ out,
                  int n_nodes, int height)
{
    __shared__ float fs[1024];
    __shared__ float pr[256];
    __shared__ float gv[16];
    __shared__ int   candsh[16];
    __shared__ int   s_cur;

    const int t = threadIdx.x;
    const int b = blockIdx.x;
    const int D = 1024;

    v4f v = *(const v4f*)(f + (size_t)b * D + t * 4);
    *(v4f*)(&fs[t * 4]) = v;
    if (t == 0) { s_cur = 0; out[b * (height + 1)] = 0; }
    __syncthreads();

    const int k = t & 15;   // candidate index
    const int g = t >> 4;   // d-group

    for (int level = 1; level <= height; ++level) {
        if (t < 16) candsh[t] = children[s_cur * 16 + t];
        __syncthreads();

        const int cand = candsh[k];
        float acc = 0.0f;
        if (cand > 0) {
            const float* xi = Xi + (cand - 1);
            #pragma unroll 8
            for (int i = 0; i < 64; ++i) {
                const int d = g + (i << 4);
                acc += xi[(size_t)d * n_nodes] * fs[d];
            }
        }
        pr[t] = acc;
        __syncthreads();

        if (t < 16) {
            float s = 0.0f;
            #pragma unroll
            for (int gg = 0; gg < 16; ++gg) s += pr[gg * 16 + t];
            gv[t] = (candsh[t] > 0) ? s : NEG_INF;
        }
        __syncthreads();

        if (t == 0) {
            int best = 0; float bg = gv[0];
            #pragma unroll
            for (int kk = 1; kk < 16; ++kk)
                if (gv[kk] > bg) { bg = gv[kk]; best = kk; }   // first-max == jnp.argmax
            bool any_valid = false;
            #pragma unroll
            for (int kk = 0; kk < 16; ++kk) any_valid |= (candsh[kk] > 0);
            const int nxt = any_valid ? candsh[best] : 0;
            out[b * (height + 1) + level] = nxt;
            if (any_valid) s_cur = nxt;
        }
        __syncthreads();
    }
}

extern "C" void kernel_launch(void* const* d_in, const int* in_sizes, int n_in,
                              void* d_out, int out_size, void* d_ws, size_t ws_size,
                              hipStream_t stream)
{
    const float* X        = (const float*)d_in[0];   // [B, K]
    const float* W        = (const float*)d_in[1];   // [K, D]
    const float* Xi       = (const float*)d_in[2];   // [D, n_nodes]
    const int*   children = (const int*)d_in[3];     // [n_nodes+1, 16]
    int*         out      = (int*)d_out;             // [B, height+1] int32

    const int D       = 1024;
    const int B       = in_sizes[0] / D;             // 4096
    const int K       = in_sizes[1] / D;             // 1024
    const int n_nodes = in_sizes[2] / D;             // 69904
    const int height  = out_size / B - 1;            // 4

    float* f = (float*)d_ws;                         // B*D f32 scratch (16 MB)

    dim3 ggrid(B / TILE_M, D / TILE_N, 1);           // 32 x 8 blocks
    gemm_f32_wmma<<<ggrid, dim3(256), 0, stream>>>(X, W, f, B, D, K);
    tree_descend<<<dim3(B), dim3(256), 0, stream>>>(f, Xi, children, out, n_nodes, height);
}